// GCN_62380105008311
// MI455X (gfx1250) — compile-verified
//
#include <hip/hip_runtime.h>
#include <hip/hip_bf16.h>

typedef float v2f __attribute__((ext_vector_type(2)));
typedef float v4f __attribute__((ext_vector_type(4)));
typedef float v8f __attribute__((ext_vector_type(8)));
typedef int   v4i __attribute__((ext_vector_type(4)));

typedef __attribute__((address_space(1))) v4i* as1_v4i_p;  // global
typedef __attribute__((address_space(3))) v4i* as3_v4i_p;  // LDS

#define BN_EPS 1e-5f
#define KT     64      // K-chunk staged in LDS
#define LDST   66      // padded LDS row stride (floats) -> avoids 64-bank conflicts

#if defined(__HIP_DEVICE_COMPILE__) && __has_builtin(__builtin_amdgcn_global_load_async_to_lds_b128)
#define HAVE_ASYNC_LDS 1
#else
#define HAVE_ASYNC_LDS 0
#endif

// ------------------------------------------------ async global -> LDS helper
__device__ __forceinline__ void copy16_to_lds(const float* g, float* l) {
#if HAVE_ASYNC_LDS
    __builtin_amdgcn_global_load_async_to_lds_b128(
        (as1_v4i_p)(v4i*)const_cast<float*>(g),
        (as3_v4i_p)(v4i*)l, 0, 0);
#else
    *(v4f*)l = *(const v4f*)g;   // ds_store_b128 fallback
#endif
}

__device__ __forceinline__ void wait_lds_fill() {
#if HAVE_ASYNC_LDS
#if __has_builtin(__builtin_amdgcn_s_wait_asynccnt)
    __builtin_amdgcn_s_wait_asynccnt(0);
#else
    asm volatile("s_wait_asynccnt 0" ::: "memory");
#endif
#endif
}

// ---------------------------------------------------------------- degree/norm
__global__ void k_deg_init(float* __restrict__ deg, int N) {
    int n = blockIdx.x * blockDim.x + threadIdx.x;
    if (n < N) deg[n] = 1.0f;  // self-loop contribution
}

__global__ void k_deg_accum(const long long* __restrict__ dst,
                            float* __restrict__ deg, int E) {
    int e = blockIdx.x * blockDim.x + threadIdx.x;
    if (e < E) atomicAdd(&deg[(int)dst[e]], 1.0f);
}

__global__ void k_dinv(const float* __restrict__ deg,
                       float* __restrict__ dinv, int N) {
    int n = blockIdx.x * blockDim.x + threadIdx.x;
    if (n < N) {
        float d = deg[n];
        dinv[n] = (d > 0.0f) ? rsqrtf(d) : 0.0f;
    }
}

// -------------------------------- pack W[K,F] into WMMA B fragments, padded.
// Bp layout: [(K+3)/4][CT][32 lanes][2] -- lane l16 carries col ct*16+l16,
// lane half hi carries K = kq*4 + 2*hi + {0,1}.  Zero-fill out of range.
__global__ void k_packB(const float* __restrict__ W, float* __restrict__ Bp,
                        int K, int F, int CT) {
    int i = blockIdx.x * blockDim.x + threadIdx.x;
    int total = ((K + 3) >> 2) * CT * 64;
    if (i >= total) return;
    int within = i & 63;
    int lane   = within >> 1;
    int j      = within & 1;
    int kq_ct  = i >> 6;
    int ct     = kq_ct % CT;
    int kq     = kq_ct / CT;
    int k   = kq * 4 + 2 * (lane >> 4) + j;
    int col = ct * 16 + (lane & 15);
    Bp[i] = (k < K && col < F) ? W[(size_t)k * F + col] : 0.0f;
}

// ------------------------------------------------------- WMMA f32 GEMM (X@W)
// Block = 4 waves (128 threads).  All waves share one 16xKT A tile in LDS;
// wave w computes column tile ct = blockIdx.x*4 + w.  Inner loop: 16
// branch-free V_WMMA_F32_16X16X4_F32 per chunk (A from LDS ds_load_b64,
// B from packed fragments, coalesced b64).  H stored with padded stride Fp.
__global__ void __launch_bounds__(128)
k_gemm_wmma(const float* __restrict__ X,   // [N, K] row-major (rows mult. of 16)
            const float* __restrict__ Bp,  // packed fragments
            float* __restrict__ H,         // [N, Fp] padded
            int N, int K, int CT, int Fp) {
    __shared__ float As[16 * LDST];
    const int tid  = threadIdx.x;
    const int wave = tid >> 5;
    const int lane = tid & 31;
    const int l16  = lane & 15;
    const int hi   = lane >> 4;
    const int m0   = blockIdx.y * 16;
    const int ct   = blockIdx.x * 4 + wave;
    const bool cact = (ct < CT);
    const int  ctc  = cact ? ct : 0;     // inactive waves use a valid dummy tile

    const int nK4 = (K + 3) >> 2;
    const float* __restrict__ aptr = As + l16 * LDST + 2 * hi;

    // cooperative-fill indexing: 128 threads, 8 floats each (two 16B chunks)
    const int fr = tid >> 3;            // row 0..15
    const int fc = (tid & 7) * 8;       // col base 0..56

    v8f acc = {};
    for (int kc = 0; kc < K; kc += KT) {
        const float* src = X + (size_t)(m0 + fr) * K + kc + fc;
        float*       dsl = As + fr * LDST + fc;
        if (kc + KT <= K) {
            copy16_to_lds(src,     dsl);
            copy16_to_lds(src + 4, dsl + 4);
            wait_lds_fill();
        } else {
#pragma unroll
            for (int j = 0; j < 8; ++j) {
                int kk = kc + fc + j;
                dsl[j] = (kk < K) ? src[j] : 0.0f;
            }
        }
        __syncthreads();

        const int kq0 = kc >> 2;
        const int kqn = min(nK4 - kq0, KT / 4);
        const float* __restrict__ bq =
            Bp + ((size_t)kq0 * CT + ctc) * 64 + lane * 2;
#pragma unroll 4
        for (int q = 0; q < kqn; ++q) {
            v2f a = *(const v2f*)(aptr + 4 * q);
            v2f b = *(const v2f*)(bq + (size_t)q * CT * 64);
            acc = __builtin_amdgcn_wmma_f32_16x16x4_f32(
                      false, a, false, b, (short)0, acc, false, false);
        }
        __syncthreads();
    }

    if (cact) {
        int col = ct * 16 + l16;
#pragma unroll
        for (int r = 0; r < 8; ++r)
            H[(size_t)(m0 + r + 8 * hi) * Fp + col] = acc[r];
    }
}

// ------------------------------------------- aggregation: bias + self-loop
__global__ void k_init_agg(const float* __restrict__ pre,  // [N, Fp]
                           const float* __restrict__ bias,
                           const float* __restrict__ dinv,
                           float* __restrict__ agg,        // [N, F] tight
                           int N, int F, int Fp) {
    int i = blockIdx.x * blockDim.x + threadIdx.x;
    if (i >= N * F) return;
    int n = i / F;
    int f = i - n * F;
    float di = dinv[n];
    agg[i] = bias[f] + pre[(size_t)n * Fp + f] * di * di;
}

// -------------------------------------- edge scatter-add (one wave per edge)
__global__ void k_scatter(const long long* __restrict__ src,
                          const long long* __restrict__ dst,
                          const float* __restrict__ dinv,
                          const float* __restrict__ pre,   // [N, Fp]
                          float* __restrict__ agg,         // [N, F]
                          int E, int F, int Fp) {
    int e    = (int)(((size_t)blockIdx.x * blockDim.x + threadIdx.x) >> 5);
    int lane = threadIdx.x & 31;
    if (e >= E) return;
    int s = (int)src[e];
    int d = (int)dst[e];
    float nrm = dinv[s] * dinv[d];
    const float* __restrict__ hs = pre + (size_t)s * Fp;
    float* __restrict__ od = agg + (size_t)d * F;
    for (int f = lane; f < F; f += 32)
        atomicAdd(&od[f], hs[f] * nrm);
}

// ------------------------------------------ BatchNorm stats over relu(agg)
__global__ void k_bnstats(const float* __restrict__ agg,
                          float* __restrict__ mu, float* __restrict__ var,
                          int N, int F) {
    __shared__ float s1[256];
    __shared__ float s2[256];
    const int c = blockIdx.x;
    const int t = threadIdx.x;
    float a = 0.0f, b = 0.0f;
    for (int n = t; n < N; n += 256) {
        float v = agg[(size_t)n * F + c];
        v = (v > 0.0f) ? v : 0.0f;  // relu before BN
        a += v;
        b += v * v;
    }
    s1[t] = a; s2[t] = b;
    __syncthreads();
    for (int o = 128; o > 0; o >>= 1) {
        if (t < o) { s1[t] += s1[t + o]; s2[t] += s2[t + o]; }
        __syncthreads();
    }
    if (t == 0) {
        float m = s1[0] / (float)N;
        mu[c]  = m;
        var[c] = s2[0] / (float)N - m * m;  // biased variance
    }
}

__global__ void k_bnapply(const float* __restrict__ agg,
                          const float* __restrict__ mu,
                          const float* __restrict__ var,
                          const float* __restrict__ g,
                          const float* __restrict__ be,
                          float* __restrict__ act, int N, int F) {
    int i = blockIdx.x * blockDim.x + threadIdx.x;
    if (i >= N * F) return;
    int f = i % F;
    float v = agg[i];
    v = (v > 0.0f) ? v : 0.0f;
    act[i] = (v - mu[f]) * rsqrtf(var[f] + BN_EPS) * g[f] + be[f];
}

// --------------------------------------------------------- final log_softmax
__global__ void k_logsoftmax(const float* __restrict__ h,
                             float* __restrict__ out, int N, int F) {
    int n = blockIdx.x * blockDim.x + threadIdx.x;
    if (n >= N) return;
    const float* __restrict__ r = h + (size_t)n * F;
    float m = -INFINITY;
    for (int f = 0; f < F; ++f) m = fmaxf(m, r[f]);
    float s = 0.0f;
    for (int f = 0; f < F; ++f) s += expf(r[f] - m);
    float l = logf(s) + m;
    for (int f = 0; f < F; ++f) out[(size_t)n * F + f] = r[f] - l;
}

// ---------------------------------------------------------------------------
extern "C" void kernel_launch(void* const* d_in, const int* in_sizes, int n_in,
                              void* d_out, int out_size, void* d_ws, size_t ws_size,
                              hipStream_t stream) {
    const float*     x  = (const float*)d_in[0];
    const long long* ei = (const long long*)d_in[1];  // int64 per reference
    const int N = in_sizes[0] / 256;
    const int E = in_sizes[1] / 2;
    const long long* src = ei;       // edge_index[0]
    const long long* dst = ei + E;   // edge_index[1]

    const float* Ws[5] = {(const float*)d_in[2], (const float*)d_in[4],
                          (const float*)d_in[6], (const float*)d_in[8],
                          (const float*)d_in[10]};
    const float* bs[5] = {(const float*)d_in[3], (const float*)d_in[5],
                          (const float*)d_in[7], (const float*)d_in[9],
                          (const float*)d_in[11]};
    const float* gs[4]  = {(const float*)d_in[12], (const float*)d_in[14],
                           (const float*)d_in[16], (const float*)d_in[18]};
    const float* bes[4] = {(const float*)d_in[13], (const float*)d_in[15],
                           (const float*)d_in[17], (const float*)d_in[19]};
    const int dims[6] = {256, 220, 150, 100, 60, 17};

    // workspace layout (floats)
    float* ws   = (float*)d_ws;
    float* deg  = ws;  ws += N;
    float* dinv = ws;  ws += N;
    float* mu   = ws;  ws += 256;
    float* var  = ws;  ws += 256;
    float* Bp   = ws;  ws += 64 * 16 * 64;       // packed B (max 64 kq x 16 ct)
    float* pre  = ws;  ws += (size_t)N * 224;    // H = X@W, padded stride
    float* agg  = ws;  ws += (size_t)N * 220;    // aggregated output, tight
    float* act  = ws;  ws += (size_t)N * 220;    // post BN+relu activation

    // symmetric GCN normalization (computed once; reused by all layers)
    k_deg_init <<<(N + 255) / 256, 256, 0, stream>>>(deg, N);
    k_deg_accum<<<(E + 255) / 256, 256, 0, stream>>>(dst, deg, E);
    k_dinv     <<<(N + 255) / 256, 256, 0, stream>>>(deg, dinv, N);

    const float* in = x;
    for (int L = 0; L < 5; ++L) {
        const int K  = dims[L];
        const int F  = dims[L + 1];
        const int CT = (F + 15) / 16;   // column tiles
        const int Fp = CT * 16;         // padded row stride of pre
        const int NF = N * F;

        int packN = ((K + 3) >> 2) * CT * 64;
        k_packB<<<(packN + 255) / 256, 256, 0, stream>>>(Ws[L], Bp, K, F, CT);

        dim3 gg((CT + 3) / 4, (N + 15) / 16);
        k_gemm_wmma<<<gg, 128, 0, stream>>>(in, Bp, pre, N, K, CT, Fp);

        k_init_agg<<<(NF + 255) / 256, 256, 0, stream>>>(pre, bs[L], dinv,
                                                         agg, N, F, Fp);
        k_scatter <<<(E + 7) / 8, 256, 0, stream>>>(src, dst, dinv, pre,
                                                    agg, E, F, Fp);

        if (L < 4) {
            k_bnstats<<<F, 256, 0, stream>>>(agg, mu, var, N, F);
            k_bnapply<<<(NF + 255) / 256, 256, 0, stream>>>(agg, mu, var,
                                                            gs[L], bes[L], act, N, F);
            in = act;
        } else {
            k_logsoftmax<<<(N + 127) / 128, 128, 0, stream>>>(agg, (float*)d_out, N, F);
        }
    }
}